// GINModel_24223615549680
// MI455X (gfx1250) — compile-verified
//
#include <hip/hip_runtime.h>

#define NN 50000
#define NE 800000
#define DIN 64
#define DH  128

typedef __attribute__((ext_vector_type(2))) float v2f;
typedef __attribute__((ext_vector_type(8))) float v8f;

__global__ void zero_kernel(float* __restrict__ p, int n) {
  int i = blockIdx.x * blockDim.x + threadIdx.x;
  if (i < n) p[i] = 0.0f;
}

__global__ void deg_kernel(const int* __restrict__ recv, float* __restrict__ deg, int E) {
  int e = blockIdx.x * blockDim.x + threadIdx.x;
  if (e < E) atomicAdd(&deg[recv[e]], 1.0f);
}

__global__ void invdeg_kernel(const float* __restrict__ deg, float* __restrict__ invd, int n) {
  int i = blockIdx.x * blockDim.x + threadIdx.x;
  if (i < n) {
    float d = deg[i];
    invd[i] = (d > 0.0f) ? (1.0f / fmaxf(d, 1.0f)) : 0.0f;
  }
}

// pooled[r,d] += h[s,d] * w[e]   (one thread per (edge, feature))
__global__ void scatter_kernel(const float* __restrict__ h, const float* __restrict__ w,
                               const int* __restrict__ snd, const int* __restrict__ rcv,
                               float* __restrict__ pooled, int E, int Dshift) {
  int i = blockIdx.x * blockDim.x + threadIdx.x;
  int total = E << Dshift;
  if (i >= total) return;
  int e = i >> Dshift;
  int d = i & ((1 << Dshift) - 1);
  int s = __ldg(&snd[e]);
  int r = __ldg(&rcv[e]);
  float we = __ldg(&w[e]);
  atomicAdd(&pooled[(r << Dshift) + d], h[(s << Dshift) + d] * we);
}

// x = h*relu(1+eps[layer]) + pooled*inv_deg ; pooled <- x (in place); optionally emb <- x
__global__ void combine_kernel(const float* __restrict__ h, float* __restrict__ pooled,
                               const float* __restrict__ invd, const float* __restrict__ eps,
                               int layer, float* __restrict__ emb, int n, int Dshift) {
  int i = blockIdx.x * blockDim.x + threadIdx.x;
  int total = n << Dshift;
  if (i >= total) return;
  int node = i >> Dshift;
  float scale = fmaxf(1.0f + __ldg(&eps[layer]), 0.0f);
  float x = h[i] * scale + pooled[i] * __ldg(&invd[node]);
  pooled[i] = x;
  if (emb) emb[i] = x;
}

// Y[M x 128] = act( X[M x K] @ W[K x 128] + bias )
// One wave computes a 16-row tile across all 8 column tiles using
// v_wmma_f32_16x16x4_f32 (full fp32). W staged into LDS once per block.
__global__ __launch_bounds__(256)
void gemm_wmma_kernel(const float* __restrict__ X, const float* __restrict__ W,
                      const float* __restrict__ bias, float* __restrict__ Y,
                      int K, int Mtiles, int doRelu) {
  __shared__ float sW[DH * DH];  // 64 KB max (K<=128, N=128)

  int tid = threadIdx.x;
  int wcount = K * DH;
  for (int i = tid; i < wcount; i += 256) sW[i] = W[i];
  __syncthreads();

  int wave = tid >> 5;
  int lane = tid & 31;
  int mtile = blockIdx.x * 8 + wave;
  if (mtile >= Mtiles) return;   // wave-uniform: whole wave skips

  int laneHi = lane >> 4;        // 0: K-pair (k0,k0+1), 1: (k0+2,k0+3)
  int lmod   = lane & 15;
  int arow   = mtile * 16 + lmod;

  v8f acc[8];
#pragma unroll
  for (int nt = 0; nt < 8; ++nt) acc[nt] = (v8f){0.f,0.f,0.f,0.f,0.f,0.f,0.f,0.f};

  for (int k0 = 0; k0 < K; k0 += 4) {
    int kk = k0 + (laneHi << 1);
    // A fragment: 16x4 fp32 per ISA layout (row striped in-lane)
    v2f a = *(const v2f*)(X + arow * K + kk);
#pragma unroll
    for (int nt = 0; nt < 8; ++nt) {
      int col = nt * 16 + lmod;
      v2f b;
      b.x = sW[kk * DH + col];
      b.y = sW[kk * DH + DH + col];
      acc[nt] = __builtin_amdgcn_wmma_f32_16x16x4_f32(
          /*neg_a=*/false, a, /*neg_b=*/false, b,
          /*c_mod=*/(short)0, acc[nt], /*reuse_a=*/false, /*reuse_b=*/false);
    }
  }

  int rowBase = mtile * 16 + (laneHi << 3);  // C/D: lanes 16-31 hold M+8
#pragma unroll
  for (int nt = 0; nt < 8; ++nt) {
    int col = nt * 16 + lmod;
    float bv = __ldg(&bias[col]);
#pragma unroll
    for (int v = 0; v < 8; ++v) {
      float val = acc[nt][v] + bv;
      if (doRelu) val = fmaxf(val, 0.0f);
      Y[(rowBase + v) * DH + col] = val;
    }
  }
}

static inline int cdiv(int a, int b) { return (a + b - 1) / b; }

extern "C" void kernel_launch(void* const* d_in, const int* in_sizes, int n_in,
                              void* d_out, int out_size, void* d_ws, size_t ws_size,
                              hipStream_t stream) {
  const float* h_in     = (const float*)d_in[0];
  const float* weights  = (const float*)d_in[1];
  const int*   senders  = (const int*)d_in[2];
  const int*   receivers= (const int*)d_in[3];
  const float* W0 = (const float*)d_in[4];
  const float* b0 = (const float*)d_in[5];
  const float* W1 = (const float*)d_in[6];
  const float* b1 = (const float*)d_in[7];
  const float* W2 = (const float*)d_in[8];
  const float* b2 = (const float*)d_in[9];
  const float* eps = (const float*)d_in[10];

  float* out       = (float*)d_out;
  float* emb_out   = out;             // node_embeddings (N x 128)
  float* final_out = out + NN * DH;   // final h (N x 128)

  // workspace: deg | inv_deg | pooled(N x 128). Layer activations ping-pong
  // through the two halves of d_out (dead until the last layer overwrites them).
  float* ws     = (float*)d_ws;
  float* deg    = ws;
  float* invd   = ws + NN;
  float* pooled = ws + 2 * NN;
  float* hA = emb_out;    // layer-0 output
  float* hB = final_out;  // layer-1 output

  const int Mtiles = NN / 16;  // 3125
  const int gemmBlocks = cdiv(Mtiles, 8);

  // degrees
  zero_kernel<<<cdiv(NN, 256), 256, 0, stream>>>(deg, NN);
  deg_kernel<<<cdiv(NE, 256), 256, 0, stream>>>(receivers, deg, NE);
  invdeg_kernel<<<cdiv(NN, 256), 256, 0, stream>>>(deg, invd, NN);

  // ---- layer 0 (D = 64) ----
  zero_kernel<<<cdiv(NN * DIN, 256), 256, 0, stream>>>(pooled, NN * DIN);
  scatter_kernel<<<cdiv(NE * DIN, 256), 256, 0, stream>>>(h_in, weights, senders, receivers, pooled, NE, 6);
  combine_kernel<<<cdiv(NN * DIN, 256), 256, 0, stream>>>(h_in, pooled, invd, eps, 0, nullptr, NN, 6);
  gemm_wmma_kernel<<<gemmBlocks, 256, 0, stream>>>(pooled, W0, b0, hA, DIN, Mtiles, 1);

  // ---- layer 1 (D = 128) ----
  zero_kernel<<<cdiv(NN * DH, 256), 256, 0, stream>>>(pooled, NN * DH);
  scatter_kernel<<<cdiv(NE * DH, 256), 256, 0, stream>>>(hA, weights, senders, receivers, pooled, NE, 7);
  combine_kernel<<<cdiv(NN * DH, 256), 256, 0, stream>>>(hA, pooled, invd, eps, 1, nullptr, NN, 7);
  gemm_wmma_kernel<<<gemmBlocks, 256, 0, stream>>>(pooled, W1, b1, hB, DH, Mtiles, 1);

  // ---- layer 2 (D = 128) ----
  zero_kernel<<<cdiv(NN * DH, 256), 256, 0, stream>>>(pooled, NN * DH);
  scatter_kernel<<<cdiv(NE * DH, 256), 256, 0, stream>>>(hB, weights, senders, receivers, pooled, NE, 7);
  combine_kernel<<<cdiv(NN * DH, 256), 256, 0, stream>>>(hB, pooled, invd, eps, 2, emb_out, NN, 7);
  gemm_wmma_kernel<<<gemmBlocks, 256, 0, stream>>>(pooled, W2, b2, final_out, DH, Mtiles, 0);
}